// HiPPO_LegS_77266461655041
// MI455X (gfx1250) — compile-verified
//
#include <hip/hip_runtime.h>

typedef __attribute__((ext_vector_type(2))) float v2f;
typedef __attribute__((ext_vector_type(8))) float v8f;

#define NSTATE 256

// ---------------------------------------------------------------------------
// Kernel A: per segment s compute
//   M_s = A_{t0+rt-1} * ... * A_{t0}   (via WMMA f32 16x16x4 matmul chain)
//   v_s = zero-state response of the segment (VALU matvec recurrence)
// M ping-pongs between two global buffers per segment (column-major:
// M[col*256 + row]).  The B-operand (Mcur) is staged 16-column blocks at a
// time into LDS with async global->LDS B128 transfers (ASYNCcnt), double
// buffered so the next block streams while WMMAs consume the current one.
// Loop order: strip (si) outer, column-block (nb) inner, so each wave keeps
// exactly ONE A-operand strip (64 x v2f = 128 VGPRs) resident -> no spills.
// ---------------------------------------------------------------------------
__global__ void __launch_bounds__(256)
hippo_segment_kernel(const float* __restrict__ A_stacked,
                     const float* __restrict__ B_stacked,
                     const float* __restrict__ inputs,
                     float* __restrict__ Mbuf,   // NSEG * 2 * N*N floats
                     float* __restrict__ vbuf,   // NSEG * N floats
                     int L, int T)
{
    const int seg  = blockIdx.x;
    const int t0   = seg * T;
    const int rt   = (T < (L - t0)) ? T : (L - t0);
    const int tid  = threadIdx.x;          // 0..255
    const int wave = tid >> 5;             // 0..7
    const int lane = tid & 31;
    const int lhalf = lane >> 4;           // 0 or 1
    const int lmod  = lane & 15;

    __shared__ float stage[2][16 * NSTATE];   // 2 x 16KB column-block buffers
    __shared__ float vsh[2][NSTATE];

    float* M0 = Mbuf + (size_t)seg * 2u * NSTATE * NSTATE;
    float* M1 = M0 + (size_t)NSTATE * NSTATE;

    // M0 = Identity (column-major), v = 0
    for (int idx = tid; idx < NSTATE * NSTATE; idx += 256) {
        int col = idx >> 8, row = idx & 255;
        M0[idx] = (row == col) ? 1.0f : 0.0f;
    }
    vsh[0][tid] = 0.0f;
    __syncthreads();

    for (int r = 0; r < rt; ++r) {
        const int t = t0 + r;
        const float* At = A_stacked + (size_t)t * NSTATE * NSTATE; // row-major
        const float u = inputs[t];
        const float* Mcur = (r & 1) ? M1 : M0;
        float*       Mnext = (r & 1) ? M0 : M1;
        const float* vcur = vsh[r & 1];

        // Async-stage one 16-column block (16 KB) of Mcur into LDS:
        // 256 threads x 4 B128 transfers, tracked by this thread's ASYNCcnt.
        auto issue_stage = [&](int nb, int buf) {
            const float* gsrc = Mcur + (size_t)nb * 16 * NSTATE + tid * 4;
            #pragma unroll
            for (int i = 0; i < 4; ++i) {
                unsigned long long ga =
                    (unsigned long long)(uintptr_t)(gsrc + i * 1024);
                unsigned la =
                    (unsigned)(uintptr_t)(&stage[buf][tid * 4 + i * 1024]);
                asm volatile("global_load_async_to_lds_b128 %0, %1, off"
                             :: "v"(la), "v"(ga) : "memory");
            }
        };

        issue_stage(0, 0);   // overlap si=0's first block with the v-matvec

        // prefetch next step's A tile while we compute
        if (r + 1 < rt)
            __builtin_prefetch(At + (size_t)NSTATE * NSTATE + (size_t)tid * NSTATE, 0, 0);

        // ---- v <- A_t v + B_t u_t  (thread tid owns row tid) ----
        float accv = 0.0f;
        {
            const float* arow = At + (size_t)tid * NSTATE;
            #pragma unroll 8
            for (int j = 0; j < NSTATE; ++j)
                accv = fmaf(arow[j], vcur[j], accv);
            accv = fmaf(B_stacked[(size_t)t * NSTATE + tid], u, accv);
        }

        // ---- Mnext = A_t * Mcur via WMMA, B-operand from LDS stage buffers.
        #pragma unroll 1
        for (int si = 0; si < 2; ++si) {
            const int i0 = (wave * 2 + si) * 16;

            // ONE A-operand strip resident (ISA 32-bit A 16x4 layout:
            // lanes 0-15 hold K=0,1; lanes 16-31 hold K=2,3).
            v2f a[64];
            const float* abase = At + (size_t)(i0 + lmod) * NSTATE + 2 * lhalf;
            #pragma unroll
            for (int k = 0; k < 64; ++k)
                a[k] = *(const v2f*)(abase + 4 * k);

            if (si == 1)
                issue_stage(0, 0);   // restart staging sweep for second strip

            #pragma unroll 1
            for (int nb = 0; nb < 16; ++nb) {
                if (nb + 1 < 16) {
                    issue_stage(nb + 1, (nb + 1) & 1);
                    asm volatile("s_wait_asynccnt 0x4" ::: "memory"); // block nb landed
                } else {
                    asm volatile("s_wait_asynccnt 0x0" ::: "memory");
                }
                __syncthreads();   // all threads' transfers for block nb visible

                // B-operand 4x16 chunk: lanes 0-15 rows 4k,4k+1; lanes 16-31 rows 4k+2,4k+3
                const float* sbase = &stage[nb & 1][lmod * NSTATE + 2 * lhalf];
                v8f acc = {};
                #pragma unroll
                for (int k = 0; k < 64; ++k) {
                    v2f b = *(const v2f*)(sbase + 4 * k);
                    acc = __builtin_amdgcn_wmma_f32_16x16x4_f32(
                        false, a[k], false, b, (short)0, acc, false, false);
                }
                // D 16x16 layout: VGPR rr holds D[rr, lane] (lanes 0-15) and
                // D[rr+8, lane-16] (lanes 16-31). Store column-major.
                float* dbase = Mnext + (size_t)(nb * 16 + lmod) * NSTATE + i0 + 8 * lhalf;
                #pragma unroll
                for (int rr = 0; rr < 8; ++rr)
                    dbase[rr] = acc[rr];

                __syncthreads();   // buffer nb&1 free for re-staging
            }
        }

        __threadfence();   // make Mnext stores visible block-wide before reuse
        __syncthreads();
        vsh[(r + 1) & 1][tid] = accv;
        __syncthreads();
    }
    vbuf[(size_t)seg * NSTATE + tid] = vsh[rt & 1][tid];
    // final M_s resides in buffer index (rt & 1): rt odd -> M1, rt even -> M0
}

// ---------------------------------------------------------------------------
// Kernel B: sequential scan over segment boundaries (1 block):
//   cbuf[0] = 0 ; cbuf[s+1] = M_s * cbuf[s] + v_s
// ---------------------------------------------------------------------------
__global__ void __launch_bounds__(256)
hippo_carry_kernel(const float* __restrict__ Mbuf,
                   const float* __restrict__ vbuf,
                   float* __restrict__ cbuf,
                   int L, int T, int NSEG)
{
    const int tid = threadIdx.x;
    __shared__ float csh[NSTATE];
    csh[tid]  = 0.0f;
    cbuf[tid] = 0.0f;
    __syncthreads();
    for (int s = 0; s < NSEG - 1; ++s) {
        const int rem = L - s * T;
        const int rt  = (T < rem) ? T : rem;
        const float* Mfin = Mbuf + ((size_t)s * 2u + (size_t)(rt & 1)) * NSTATE * NSTATE;
        float acc = 0.0f;
        // column-major M: for fixed col, threads read consecutive rows (coalesced)
        for (int col = 0; col < NSTATE; ++col)
            acc = fmaf(Mfin[(size_t)col * NSTATE + tid], csh[col], acc);
        acc += vbuf[(size_t)s * NSTATE + tid];
        __syncthreads();
        csh[tid] = acc;
        cbuf[(size_t)(s + 1) * NSTATE + tid] = acc;
        __syncthreads();
    }
}

// ---------------------------------------------------------------------------
// Kernel C: replay each segment from its carried-in state; emit predictions.
// ---------------------------------------------------------------------------
__global__ void __launch_bounds__(256)
hippo_emit_kernel(const float* __restrict__ A_stacked,
                  const float* __restrict__ B_stacked,
                  const float* __restrict__ C_discr,
                  const float* __restrict__ D_discr,
                  const float* __restrict__ inputs,
                  const float* __restrict__ cbuf,
                  float* __restrict__ out,
                  int L, int T)
{
    const int seg = blockIdx.x;
    const int t0  = seg * T;
    const int rt  = (T < (L - t0)) ? T : (L - t0);
    const int tid = threadIdx.x;
    __shared__ float csh[2][NSTATE];
    __shared__ float red[8];
    csh[0][tid] = cbuf[(size_t)seg * NSTATE + tid];
    __syncthreads();

    for (int r = 0; r < rt; ++r) {
        const int t = t0 + r;
        const float* At = A_stacked + (size_t)t * NSTATE * NSTATE;
        const float u = inputs[t];
        const float* cc = csh[r & 1];

        if (r + 1 < rt)
            __builtin_prefetch(At + (size_t)NSTATE * NSTATE + (size_t)tid * NSTATE, 0, 0);

        float acc = 0.0f;
        const float* arow = At + (size_t)tid * NSTATE;
        #pragma unroll 8
        for (int j = 0; j < NSTATE; ++j)
            acc = fmaf(arow[j], cc[j], acc);
        acc = fmaf(B_stacked[(size_t)t * NSTATE + tid], u, acc);

        // pred_t = f_t * dot(c_t, C_t) + D_t * u_t  (wave32 shuffle + LDS reduce)
        float p = acc * C_discr[(size_t)t * NSTATE + tid];
        #pragma unroll
        for (int off = 16; off > 0; off >>= 1)
            p += __shfl_down(p, off, 32);
        if ((tid & 31) == 0) red[tid >> 5] = p;
        __syncthreads();
        if (tid == 0) {
            float s = 0.0f;
            #pragma unroll
            for (int w = 0; w < 8; ++w) s += red[w];
            const float f = (t == 0) ? 1.0f : 2.0f;
            out[t] = fmaf(f, s, D_discr[t] * u);
        }
        csh[(r + 1) & 1][tid] = acc;
        __syncthreads();
    }
}

__global__ void hippo_zero_kernel(float* p, int n)
{
    int i = blockIdx.x * blockDim.x + threadIdx.x;
    if (i < n) p[i] = 0.0f;
}

// ---------------------------------------------------------------------------
extern "C" void kernel_launch(void* const* d_in, const int* in_sizes, int n_in,
                              void* d_out, int out_size, void* d_ws, size_t ws_size,
                              hipStream_t stream)
{
    const float* inputs = (const float*)d_in[0];
    const float* A      = (const float*)d_in[1];
    const float* B      = (const float*)d_in[2];
    const float* C      = (const float*)d_in[3];
    const float* D      = (const float*)d_in[4];
    float* out = (float*)d_out;
    const int L = in_sizes[0];
    if (L <= 0) return;

    // Pick segment length T so ws holds: M ping-pong (NSEG*2*N*N) + v + c_in.
    int T = 16;
    size_t nseg = 0, need = 0;
    for (;;) {
        nseg = ((size_t)L + T - 1) / (size_t)T;
        need = (nseg * 2u * NSTATE * NSTATE + 2u * nseg * NSTATE) * sizeof(float);
        if (need <= ws_size || T >= L) break;
        T <<= 1;
    }

    if (need <= ws_size) {
        float* Mbuf = (float*)d_ws;
        float* vbuf = Mbuf + nseg * 2u * NSTATE * NSTATE;
        float* cbuf = vbuf + nseg * NSTATE;

        hipLaunchKernelGGL(hippo_segment_kernel, dim3((unsigned)nseg), dim3(256), 0, stream,
                           A, B, inputs, Mbuf, vbuf, L, T);
        hipLaunchKernelGGL(hippo_carry_kernel, dim3(1), dim3(256), 0, stream,
                           Mbuf, vbuf, cbuf, L, T, (int)nseg);
        hipLaunchKernelGGL(hippo_emit_kernel, dim3((unsigned)nseg), dim3(256), 0, stream,
                           A, B, C, D, inputs, cbuf, out, L, T);
    } else if (ws_size >= NSTATE * sizeof(float)) {
        // Fallback: tiny scratch -> single-block fully sequential scan (no WMMA).
        float* cbuf = (float*)d_ws;
        hipLaunchKernelGGL(hippo_zero_kernel, dim3(1), dim3(256), 0, stream, cbuf, NSTATE);
        hipLaunchKernelGGL(hippo_emit_kernel, dim3(1), dim3(256), 0, stream,
                           A, B, C, D, inputs, cbuf, out, L, L);
    }
}